// DimensionalConsistencyLoss_22247930593476
// MI455X (gfx1250) — compile-verified
//
#include <hip/hip_runtime.h>
#include <hip/hip_bf16.h>

typedef float v2f __attribute__((ext_vector_type(2)));
typedef float v8f __attribute__((ext_vector_type(8)));

#define DIM 512
#define N_POS 4096
#define N_NEG 4096
#define N_NEU 2048
#define NUM_CONSTRAINTS (N_POS + N_NEG + N_NEU)

#define THREADS 256
#define BLOCKS 64
#define WAVES_PER_BLOCK (THREADS / 32)
#define ROWS_PER_WAVE 16
#define POS_GROUPS (N_POS / ROWS_PER_WAVE) /* 256 */
#define SPARSITY_PER (0.1f / (float)(DIM - 1))

// Each wave reduces 16 gathered embedding rows with V_WMMA_F32_16X16X4_F32:
//   D = |A| x ones + C  -> every column of D holds the 16 row abs-sums.
// A-layout (ISA 7.12.2, 32-bit A 16x4): lanes 0-15 provide K=0,1; lanes 16-31
// provide K=2,3 for the same M=lane%16 row. B = all-ones is layout-invariant.
// D-layout: VGPR r @ lane 0 = D[r][0], VGPR r @ lane 16 = D[8+r][0].
__global__ __launch_bounds__(THREADS) void dcl_main_kernel(
    const float* __restrict__ emb,
    const int* __restrict__ pos_ids, const int* __restrict__ pos_dims,
    const int* __restrict__ neg_ids, const int* __restrict__ neg_dims,
    const int* __restrict__ neu_ids, const int* __restrict__ neu_dims,
    float* __restrict__ block_partials)
{
    const int lane  = threadIdx.x & 31;
    const int wave  = threadIdx.x >> 5;
    const int gwave = blockIdx.x * WAVES_PER_BLOCK + wave; // 0..511, one group each

    float partial = 0.0f;

    // ---- pos/neg constraint group (wave-uniform selection; EXEC all ones) ----
    {
        const bool isPos = (gwave < POS_GROUPS);
        const int  g     = isPos ? gwave : (gwave - POS_GROUPS);
        const int* ids   = isPos ? pos_ids  : neg_ids;
        const int* dims  = isPos ? pos_dims : neg_dims;
        const int  base  = g * ROWS_PER_WAVE;

        const int    r      = lane & 15;
        const int    row    = ids[base + r];
        const float* rowPtr = emb + (size_t)row * DIM;
        const int    koff   = (lane >> 4) * 2; // lanes 0-15 -> K 0,1 ; lanes 16-31 -> K 2,3

        v2f bOnes; bOnes.x = 1.0f; bOnes.y = 1.0f;
        v8f acc = {};
        #pragma unroll 4
        for (int k = 0; k < DIM; k += 4) {
            if ((k & 63) == 0)
                __builtin_prefetch(rowPtr + k + 64, 0, 3); // global_prefetch_b8
            v2f av = *(const v2f*)(rowPtr + k + koff);     // 8B gather load per lane
            v2f a;
            a.x = __builtin_fabsf(av.x);
            a.y = __builtin_fabsf(av.y);
            acc = __builtin_amdgcn_wmma_f32_16x16x4_f32(
                /*neg_a=*/false, a, /*neg_b=*/false, bOnes,
                /*c_mod=*/(short)0, acc, /*reuse_a=*/false, /*reuse_b=*/false);
        }

        // Column N==0 of D: lane 0 holds rows 0..7, lane 16 holds rows 8..15.
        float colsum = acc[0] + acc[1] + acc[2] + acc[3] +
                       acc[4] + acc[5] + acc[6] + acc[7];
        if (lane == 0 || lane == 16)
            partial += SPARSITY_PER * colsum;

        // Sign loss + |t| sparsity correction: one constraint per lane (0..15).
        if (lane < 16) {
            const int   d  = dims[base + lane];
            const float t  = rowPtr[d];
            const float at = __builtin_fabsf(t);
            float sl;
            if (isPos)
                sl = (t <= 0.0f) ? (at + 0.1f) : (-0.1f * t);
            else
                sl = (t >= 0.0f) ? (at + 0.1f) : (-0.1f * at);
            partial += sl - SPARSITY_PER * at;
        }
    }

    // ---- neutral constraints: first 2048 global threads, one each ----
    {
        const int tid = blockIdx.x * THREADS + threadIdx.x;
        if (tid < N_NEU) {
            const int   row = neu_ids[tid];
            const int   d   = neu_dims[tid];
            const float t   = emb[(size_t)row * DIM + d];
            partial += 2.0f * __builtin_fabsf(t);
        }
    }

    // ---- deterministic in-block tree reduction (fixed order) ----
    __shared__ float sdata[THREADS];
    sdata[threadIdx.x] = partial;
    __syncthreads();
    #pragma unroll
    for (int s = THREADS / 2; s > 0; s >>= 1) {
        if (threadIdx.x < (unsigned)s)
            sdata[threadIdx.x] += sdata[threadIdx.x + s];
        __syncthreads();
    }
    if (threadIdx.x == 0)
        block_partials[blockIdx.x] = sdata[0];
}

// Fixed-order final fold of the 64 block partials -> scalar (deterministic).
__global__ void dcl_final_kernel(const float* __restrict__ block_partials,
                                 float* __restrict__ out)
{
    if (threadIdx.x == 0) {
        float s = 0.0f;
        #pragma unroll
        for (int i = 0; i < BLOCKS; ++i)
            s += block_partials[i];
        out[0] = 0.5f * (s / (float)NUM_CONSTRAINTS);
    }
}

extern "C" void kernel_launch(void* const* d_in, const int* in_sizes, int n_in,
                              void* d_out, int out_size, void* d_ws, size_t ws_size,
                              hipStream_t stream) {
    const float* emb      = (const float*)d_in[0];
    const int*   pos_ids  = (const int*)d_in[1];
    const int*   pos_dims = (const int*)d_in[2];
    const int*   neg_ids  = (const int*)d_in[3];
    const int*   neg_dims = (const int*)d_in[4];
    const int*   neu_ids  = (const int*)d_in[5];
    const int*   neu_dims = (const int*)d_in[6];
    float*       partials = (float*)d_ws;   // 64 floats of scratch
    float*       out      = (float*)d_out;

    dcl_main_kernel<<<BLOCKS, THREADS, 0, stream>>>(
        emb, pos_ids, pos_dims, neg_ids, neg_dims, neu_ids, neu_dims, partials);
    dcl_final_kernel<<<1, 32, 0, stream>>>(partials, out);
}